// DistanceLossFast_63582695850674
// MI455X (gfx1250) — compile-verified
//
#include <hip/hip_runtime.h>
#include <math.h>

#define STEPS   200
#define DT      0.0002f
#define MAXN    8192
#define BLOCK   1024
#define RPT     8          // rays/thread in generic fallback
#define NB      16         // blocks in the halo fast path
#define CHUNK   512        // owned rays per block (NB*CHUNK == 8192)
#define EXT     (CHUNK + 2 * STEPS)   // 912: owned + halo on each side

// ---------------------------------------------------------------------------
// gfx1250 split workgroup barrier that orders ONLY LDS traffic (s_wait_dscnt),
// leaving global loads in flight across the barrier (they are tracked by the
// separate LOADcnt on CDNA5 and only feed this thread's own VGPRs).
// ---------------------------------------------------------------------------
__device__ __forceinline__ void wg_barrier_lds() {
  asm volatile("s_wait_dscnt 0\n\t"
               "s_barrier_signal -1\n\t"
               "s_barrier_wait -1" ::: "memory");
}

// ---------------------------------------------------------------------------
// Prologue: interleave the 4 sampled channels (data0, data1, beta, kappa)
// into one float4 plane so each bilinear corner is one global_load_b128.
// ---------------------------------------------------------------------------
__global__ void repack_kernel(const float* __restrict__ data,
                              const float* __restrict__ beta,
                              const float* __restrict__ kappa,
                              float4* __restrict__ packed, int HW) {
  int i = blockIdx.x * blockDim.x + threadIdx.x;
  if (i < HW) {
    float4 v;
    v.x = data[i];        // data channel 0
    v.y = data[i + HW];   // data channel 1
    v.z = beta[i];
    v.w = kappa[i];
    packed[i] = v;
  }
}

// In-flight bilinear sample: 4 corner float4 loads + interpolation weights.
struct Pend {
  float4 A, B, C, D;
  float wx, wy;
};

// Issue the 4 corner global_load_b128 for position derived from r.
// W compile-time: 3 corners use immediate ioffsets off one address VGPR.
template<int W>
__device__ __forceinline__ void issue_sample(const float4* __restrict__ packed,
                                             float r, float ct, float st,
                                             float ox, float oy, float cmax,
                                             Pend& p) {
  float x = ox + r * ct;
  float y = oy + r * st;
  x = fminf(fmaxf(x, 0.0f), cmax);
  y = fminf(fmaxf(y, 0.0f), cmax);
  float xf = floorf(x), yf = floorf(y);
  p.wx = x - xf;
  p.wy = y - yf;
  int idx = (int)yf * W + (int)xf;
  p.A = packed[idx];          // (x0,y0)
  p.B = packed[idx + 1];      // (x1,y0)
  p.C = packed[idx + W];      // (x0,y1)
  p.D = packed[idx + W + 1];  // (x1,y1)
}

// Consume an in-flight sample: 4-channel bilinear (float4 math -> pk fma).
__device__ __forceinline__ float4 consume_sample(const Pend& p) {
  float owx = 1.0f - p.wx;
  float owy = 1.0f - p.wy;
  float4 top = p.A * owx + p.B * p.wx;
  float4 bot = p.C * owx + p.D * p.wx;
  return top * owy + bot * p.wy;   // (g0, g1, beta, kappa)
}

// ---------------------------------------------------------------------------
// Fast path: redundant-halo decomposition, 16 blocks x 1024 threads on 16
// WGPs, NO inter-block synchronization. Block b owns global rays
// [b*CHUNK,(b+1)*CHUNK); local l <-> global (b*CHUNK - STEPS + l) mod N.
// State s valid on local [s, EXT-s); after 200 steps owned rays are exact.
// One ray per thread in a register; neighbors via double-buffered LDS.
// The next step's 4 corner gathers are issued BEFORE the barrier (position
// depends only on the thread's own rn), and the barrier waits only dscnt,
// so the L2 round trip overlaps the halo exchange + barrier + next step's
// LDS reads instead of serializing with them.
// ---------------------------------------------------------------------------
template<int WDIM>
__launch_bounds__(BLOCK)
__global__ void evolve_halo(const float* __restrict__ rho_init,
                            const float* __restrict__ rho_target,
                            const float* __restrict__ origin,
                            const float* __restrict__ theta,
                            const float* __restrict__ dtheta_p,
                            const float4* __restrict__ packed,
                            float* __restrict__ out,
                            float* __restrict__ partials,
                            int N) {
  __shared__ float buf[2 * EXT];   // double buffer, indexed cur*EXT + l

  const int t    = threadIdx.x;
  const int b    = blockIdx.x;
  const int g0   = b * CHUNK;
  const int mask = N - 1;          // N == 8192 (power of two)

  const float ox  = origin[0];
  const float oy  = origin[1];
  const float dth = dtheta_p[0];
  const float inv_dt2 = 1.0f / (dth * dth);
  const float rho_max = 0.5f * (float)WDIM - 2.0f;
  const float cmax    = (float)WDIM - 1.001f;

  // Thread t owns local ray l = t (t < EXT); others only help reduce.
  const int g = (g0 - STEPS + t + N) & mask;   // global ray id for local t
  float r = 0.0f, ct = 0.0f, st = 0.0f;
  if (t < EXT) {
    r = rho_init[g];
    buf[t] = r;                     // state 0 into buffer 0
    sincosf(theta[g], &st, &ct);
  }
  __builtin_prefetch(rho_target + g, 0, 1);    // global_prefetch_b8

  // Thread t is active for steps s < lim (shrinking valid window).
  const int lim = min(t, EXT - 1 - t);         // <=0 for edges / t >= EXT

  // Pipeline prologue: issue step-0 gathers; latency overlaps init barrier.
  Pend p;
  if (0 < lim) issue_sample<WDIM>(packed, r, ct, st, ox, oy, cmax, p);
  wg_barrier_lds();

  int cur = 0;
  for (int s = 0; s < STEPS; ++s) {
    float rn = r;
    if (s < lim) {
      float rm = buf[cur * EXT + t - 1];
      float rp = buf[cur * EXT + t + 1];
      float4 smp = consume_sample(p);          // waits on in-flight loads
      float radial = smp.x * ct + smp.y * st;
      float lap    = (rm - 2.0f * r + rp) * inv_dt2;
      float v      = r + DT * (-radial + smp.z * lap + smp.w);
      rn = fminf(fmaxf(v, 1.0f), rho_max);
    }
    // Issue next step's gathers NOW: position needs only this thread's rn,
    // so the L2 round trip overlaps the halo store + barrier below.
    if (s + 1 < lim) issue_sample<WDIM>(packed, rn, ct, st, ox, oy, cmax, p);

    int nxt = cur ^ 1;
    if (s < lim) buf[nxt * EXT + t] = rn;
    r = rn;
    wg_barrier_lds();               // waits dscnt only; gathers stay in flight
    cur = nxt;
  }

  // Epilogue: owned rays are local [STEPS, STEPS+CHUNK).
  float partial = 0.0f;
  if (t >= STEPS && t < STEPS + CHUNK) {
    int gi = g;                     // == g0 + (t - STEPS), already in [0,N)
    out[1 + 2 * gi]     = ox + r * ct;
    out[1 + 2 * gi + 1] = oy + r * st;
    out[1 + 2 * N + gi] = r;
    partial = fabsf(r - rho_target[gi]);
  }

  // Block-level reduction of |rho - target| into partials[b].
  float* red = &buf[0];             // 2*EXT = 1824 floats >= BLOCK
  __syncthreads();                  // full fence: re-using buf for reduction
  red[t] = partial;
  __syncthreads();
  for (int off = BLOCK >> 1; off > 0; off >>= 1) {
    if (t < off) red[t] += red[t + off];
    __syncthreads();
  }
  if (t == 0) partials[b] = red[0];
}

// Final: sum the NB block partials -> mean abs diff.
__global__ void finish_kernel(const float* __restrict__ partials,
                              float* __restrict__ out, int N) {
  if (blockIdx.x == 0 && threadIdx.x == 0) {
    float s = 0.0f;
    for (int i = 0; i < NB; ++i) s += partials[i];
    out[0] = s / (float)N;
  }
}

// ---------------------------------------------------------------------------
// Generic fallback (any N <= MAXN, any H/W, packed or planar gathers).
// ---------------------------------------------------------------------------
__device__ __forceinline__ void bilinear_gen(const float4* __restrict__ packed,
                                             int usePacked,
                                             const float* __restrict__ data,
                                             const float* __restrict__ beta,
                                             const float* __restrict__ kappa,
                                             int HW, int W, float x, float y,
                                             float& g0, float& g1, float& bb, float& kk) {
  float xf = floorf(x), yf = floorf(y);
  int   x0 = (int)xf,   y0 = (int)yf;
  float wx = x - xf,    wy = y - yf;
  float owx = 1.0f - wx, owy = 1.0f - wy;
  long idx = (long)y0 * (long)W + (long)x0;
  if (usePacked) {
    float4 a = packed[idx], b = packed[idx + 1];
    float4 c = packed[idx + W], d = packed[idx + W + 1];
    float t0 = a.x * owx + b.x * wx, b0 = c.x * owx + d.x * wx;
    float t1 = a.y * owx + b.y * wx, b1 = c.y * owx + d.y * wx;
    float t2 = a.z * owx + b.z * wx, b2 = c.z * owx + d.z * wx;
    float t3 = a.w * owx + b.w * wx, b3 = c.w * owx + d.w * wx;
    g0 = t0 * owy + b0 * wy;  g1 = t1 * owy + b1 * wy;
    bb = t2 * owy + b2 * wy;  kk = t3 * owy + b3 * wy;
  } else {
    const float* d0 = data;
    const float* d1 = data + HW;
    float tp, bt;
    tp = d0[idx] * owx + d0[idx + 1] * wx;
    bt = d0[idx + W] * owx + d0[idx + W + 1] * wx;
    g0 = tp * owy + bt * wy;
    tp = d1[idx] * owx + d1[idx + 1] * wx;
    bt = d1[idx + W] * owx + d1[idx + W + 1] * wx;
    g1 = tp * owy + bt * wy;
    tp = beta[idx] * owx + beta[idx + 1] * wx;
    bt = beta[idx + W] * owx + beta[idx + W + 1] * wx;
    bb = tp * owy + bt * wy;
    tp = kappa[idx] * owx + kappa[idx + 1] * wx;
    bt = kappa[idx + W] * owx + kappa[idx + W + 1] * wx;
    kk = tp * owy + bt * wy;
  }
}

__launch_bounds__(BLOCK)
__global__ void evolve_generic(const float* __restrict__ rho_init,
                               const float* __restrict__ rho_target,
                               const float* __restrict__ origin,
                               const float* __restrict__ beta,
                               const float* __restrict__ data,
                               const float* __restrict__ kappa,
                               const float* __restrict__ theta,
                               const float* __restrict__ dtheta_p,
                               const float4* __restrict__ packed,
                               int usePacked,
                               float* __restrict__ out,
                               int N, int H, int W) {
  __shared__ float buf[2][MAXN];
  const int tid = threadIdx.x;
  const int T   = BLOCK;
  const int HW  = H * W;

  const float ox  = origin[0];
  const float oy  = origin[1];
  const float dth = dtheta_p[0];
  const float inv_dt2 = 1.0f / (dth * dth);
  const float rho_max = 0.5f * (float)(H < W ? H : W) - 2.0f;
  const float xmax = (float)W - 1.001f;
  const float ymax = (float)H - 1.001f;

  float ct[RPT], st[RPT];
  for (int j = 0; j < RPT; ++j) {
    int i = tid + j * T;
    if (i >= N) break;
    float s, c;
    sincosf(theta[i], &s, &c);
    ct[j] = c; st[j] = s;
    buf[0][i] = rho_init[i];
  }
  __syncthreads();

  int cur = 0;
  for (int step = 0; step < STEPS; ++step) {
    float rnew[RPT];
    for (int j = 0; j < RPT; ++j) {
      int i = tid + j * T;
      if (i >= N) break;
      float r  = buf[cur][i];
      float rm = buf[cur][(i == 0) ? (N - 1) : (i - 1)];
      float rp = buf[cur][(i == N - 1) ? 0 : (i + 1)];
      float x = ox + r * ct[j];
      float y = oy + r * st[j];
      x = fminf(fmaxf(x, 0.0f), xmax);
      y = fminf(fmaxf(y, 0.0f), ymax);
      float g0, g1, bb, kk;
      bilinear_gen(packed, usePacked, data, beta, kappa, HW, W, x, y, g0, g1, bb, kk);
      float radial = g0 * ct[j] + g1 * st[j];
      float lap    = (rm - 2.0f * r + rp) * inv_dt2;
      float v      = r + DT * (-radial + bb * lap + kk);
      rnew[j] = fminf(fmaxf(v, 1.0f), rho_max);
    }
    int nxt = cur ^ 1;
    for (int j = 0; j < RPT; ++j) {
      int i = tid + j * T;
      if (i >= N) break;
      buf[nxt][i] = rnew[j];
    }
    __syncthreads();
    cur = nxt;
  }

  float partial = 0.0f;
  for (int j = 0; j < RPT; ++j) {
    int i = tid + j * T;
    if (i >= N) break;
    float r = buf[cur][i];
    out[1 + 2 * i]     = ox + r * ct[j];
    out[1 + 2 * i + 1] = oy + r * st[j];
    out[1 + 2 * N + i] = r;
    partial += fabsf(r - rho_target[i]);
  }
  float* red = &buf[cur ^ 1][0];
  red[tid] = partial;
  __syncthreads();
  for (int off = T >> 1; off > 0; off >>= 1) {
    if (tid < off) red[tid] += red[tid + off];
    __syncthreads();
  }
  if (tid == 0) out[0] = red[0] / (float)N;
}

// ---------------------------------------------------------------------------
extern "C" void kernel_launch(void* const* d_in, const int* in_sizes, int n_in,
                              void* d_out, int out_size, void* d_ws, size_t ws_size,
                              hipStream_t stream) {
  const float* rho_init   = (const float*)d_in[0];
  const float* rho_target = (const float*)d_in[1];
  const float* origin     = (const float*)d_in[2];
  const float* beta       = (const float*)d_in[3];
  const float* data       = (const float*)d_in[4];
  const float* kappa      = (const float*)d_in[5];
  const float* theta      = (const float*)d_in[6];
  const float* dtheta     = (const float*)d_in[7];

  int N  = in_sizes[0];
  if (N > MAXN) N = MAXN;          // reference uses N = 8192
  int HW = in_sizes[3];            // beta is H*W, square maps per reference
  int H  = (int)llround(sqrt((double)HW));
  int W  = H;

  size_t packedBytes = (size_t)HW * sizeof(float4);
  float4* packed = (float4*)d_ws;
  float*  partials = (float*)((char*)d_ws + packedBytes);
  int usePacked = (ws_size >= packedBytes) ? 1 : 0;

  if (usePacked) {
    repack_kernel<<<(HW + 255) / 256, 256, 0, stream>>>(data, beta, kappa, packed, HW);
  }

  bool fast = usePacked && (N == NB * CHUNK) && (W == 2048) && (H == 2048) &&
              (ws_size >= packedBytes + NB * sizeof(float));
  if (fast) {
    evolve_halo<2048><<<NB, BLOCK, 0, stream>>>(rho_init, rho_target, origin,
                                                theta, dtheta, packed,
                                                (float*)d_out, partials, N);
    finish_kernel<<<1, 32, 0, stream>>>(partials, (float*)d_out, N);
  } else {
    evolve_generic<<<1, BLOCK, 0, stream>>>(rho_init, rho_target, origin, beta,
                                            data, kappa, theta, dtheta, packed,
                                            usePacked, (float*)d_out, N, H, W);
  }
}